// RGCNEncoder_20641612825459
// MI455X (gfx1250) — compile-verified
//
#include <hip/hip_runtime.h>

// ---------------------------------------------------------------------------
// RGCN encoder for MI455X (gfx1250, wave32, WMMA).
//
// Layer: out = (mean_{r} agg) @ W_r  +  x @ root + bias   (+ ReLU after L1)
// Strategy:
//   K1: W_r = sum_b comp[r,b] * basis[b]            (tiny)
//   K2: zero agg[N,R,128] + cnt[N,R]                (float4 stores)
//   K3: per-edge wave32 scatter: atomicAdd x[src] into agg[dst,r], cnt++
//   K4: WMMA transform: per wave, 16-node x 32-col tile, 9 GEMM terms
//       using V_WMMA_F32_16X16X4_F32 (full f32 precision, matches reference)
// ---------------------------------------------------------------------------

typedef __attribute__((ext_vector_type(2))) float v2f;
typedef __attribute__((ext_vector_type(8))) float v8f;

#define RGCN_D   128
#define RGCN_R   8
#define RGCN_B   8
#define RGCN_DD  (RGCN_D * RGCN_D)

// ---- K1: W[r,i,o] = sum_b comp[r,b] * basis[b,i,o] ------------------------
__global__ __launch_bounds__(256) void rgcn_compute_w(
    const float* __restrict__ comp,   // [R,B]
    const float* __restrict__ basis,  // [B,D,D]
    float* __restrict__ W)            // [R,D,D]
{
    int idx = blockIdx.x * blockDim.x + threadIdx.x;   // over R*D*D
    if (idx >= RGCN_R * RGCN_DD) return;
    int r  = idx / RGCN_DD;
    int io = idx - r * RGCN_DD;
    float s = 0.f;
#pragma unroll
    for (int b = 0; b < RGCN_B; ++b)
        s += comp[r * RGCN_B + b] * basis[(size_t)b * RGCN_DD + io];
    W[idx] = s;
}

// ---- K2: zero a float buffer (float4 vectorized, grid-stride) -------------
__global__ __launch_bounds__(256) void rgcn_zero_f32(float4* __restrict__ p, long n4)
{
    long i = (long)blockIdx.x * blockDim.x + threadIdx.x;
    long stride = (long)gridDim.x * blockDim.x;
    float4 z = make_float4(0.f, 0.f, 0.f, 0.f);
    for (; i < n4; i += stride) p[i] = z;
}

// ---- K3: scatter-aggregate. One wave32 per edge: 32 lanes x float4 = 128 --
__global__ __launch_bounds__(256) void rgcn_scatter(
    const float* __restrict__ x,    // [N,128]
    const int*   __restrict__ src,  // [E]
    const int*   __restrict__ dst,  // [E]
    const int*   __restrict__ et,   // [E]
    float* __restrict__ agg,        // [N,R,128]
    float* __restrict__ cnt,        // [N,R]
    int E)
{
    int e    = (blockIdx.x * blockDim.x + threadIdx.x) >> 5;
    int lane = threadIdx.x & 31;
    if (e >= E) return;
    int s = src[e], d = dst[e], r = et[e];
    const float4* xs = (const float4*)(x + (size_t)s * RGCN_D);
    float4 v = xs[lane];
    float* ap = agg + ((size_t)d * RGCN_R + r) * RGCN_D + lane * 4;
    atomicAdd(ap + 0, v.x);
    atomicAdd(ap + 1, v.y);
    atomicAdd(ap + 2, v.z);
    atomicAdd(ap + 3, v.w);
    if (lane == 0) atomicAdd(cnt + (size_t)d * RGCN_R + r, 1.0f);
}

// ---- K4: WMMA transform ---------------------------------------------------
// One wave per 16-node x 32-col output tile (two 16x16 f32 accumulators).
// A-frag (16x4 f32): lane half g=lane>>4 holds K=2g,2g+1 of row lane&15.
// B-frag (4x16 f32): lane half g holds rows K=2g,2g+1 at col lane&15.
// C/D (16x16 f32):   VGPR v = row v+8g, col lane&15.
__global__ __launch_bounds__(32) void rgcn_transform(
    const float* __restrict__ xin,   // [N,128] layer input
    const float* __restrict__ agg,   // [N,R,128]
    const float* __restrict__ cnt,   // [N,R]
    const float* __restrict__ Wrel,  // [R,128,128]
    const float* __restrict__ root,  // [128,128]
    const float* __restrict__ bias,  // [128]
    float* __restrict__ outp,        // [N,128]
    int relu)
{
    const int lane   = threadIdx.x;        // 0..31
    const int g      = lane >> 4;          // half-wave group
    const int m      = lane & 15;          // A-row / C-col within tile
    const int ctiles = RGCN_D / 32;        // 4 column-tile pairs
    const int n0 = (blockIdx.x / ctiles) * 16;
    const int c0 = (blockIdx.x % ctiles) * 32;

    // bias epilogue folded into initial accumulators (col depends on lane only)
    float b0 = bias[c0 + m];
    float b1 = bias[c0 + 16 + m];
    v8f acc0, acc1;
#pragma unroll
    for (int v = 0; v < 8; ++v) { acc0[v] = b0; acc1[v] = b1; }

    // ---- root term: A = xin rows ----
    {
        const float* arow = xin + (size_t)(n0 + m) * RGCN_D;
#pragma unroll 4
        for (int k0 = 0; k0 < RGCN_D; k0 += 4) {
            int k = k0 + 2 * g;
            v2f a;  a.x = arow[k];            a.y = arow[k + 1];
            v2f bA; bA.x = root[k * RGCN_D + c0 + m];
                    bA.y = root[(k + 1) * RGCN_D + c0 + m];
            v2f bB; bB.x = root[k * RGCN_D + c0 + 16 + m];
                    bB.y = root[(k + 1) * RGCN_D + c0 + 16 + m];
            acc0 = __builtin_amdgcn_wmma_f32_16x16x4_f32(
                false, a, false, bA, (short)0, acc0, false, false);
            acc1 = __builtin_amdgcn_wmma_f32_16x16x4_f32(
                false, a, false, bB, (short)0, acc1, false, false);
        }
    }

    // ---- per-relation mean term: A = agg rows scaled by 1/max(cnt,1) ----
#pragma unroll 1
    for (int r = 0; r < RGCN_R; ++r) {
        float c   = cnt[(size_t)(n0 + m) * RGCN_R + r];
        float inv = 1.0f / fmaxf(c, 1.0f);
        const float* grow = agg + ((size_t)(n0 + m) * RGCN_R + r) * RGCN_D;
        const float* wr   = Wrel + (size_t)r * RGCN_DD;
#pragma unroll 4
        for (int k0 = 0; k0 < RGCN_D; k0 += 4) {
            int k = k0 + 2 * g;
            v2f a;  a.x = grow[k] * inv;      a.y = grow[k + 1] * inv;
            v2f bA; bA.x = wr[k * RGCN_D + c0 + m];
                    bA.y = wr[(k + 1) * RGCN_D + c0 + m];
            v2f bB; bB.x = wr[k * RGCN_D + c0 + 16 + m];
                    bB.y = wr[(k + 1) * RGCN_D + c0 + 16 + m];
            acc0 = __builtin_amdgcn_wmma_f32_16x16x4_f32(
                false, a, false, bA, (short)0, acc0, false, false);
            acc1 = __builtin_amdgcn_wmma_f32_16x16x4_f32(
                false, a, false, bB, (short)0, acc1, false, false);
        }
    }

    // ---- store (fused ReLU for layer 1) ----
#pragma unroll
    for (int v = 0; v < 8; ++v) {
        float o0 = acc0[v], o1 = acc1[v];
        if (relu) { o0 = fmaxf(o0, 0.f); o1 = fmaxf(o1, 0.f); }
        size_t row = (size_t)(n0 + v + 8 * g) * RGCN_D;
        outp[row + c0 + m]      = o0;
        outp[row + c0 + 16 + m] = o1;
    }
}

// ---------------------------------------------------------------------------
extern "C" void kernel_launch(void* const* d_in, const int* in_sizes, int n_in,
                              void* d_out, int out_size, void* d_ws, size_t ws_size,
                              hipStream_t stream)
{
    const float* x      = (const float*)d_in[0];   // [N,128]
    const int*   eidx   = (const int*)  d_in[1];   // [2,E]
    const int*   etype  = (const int*)  d_in[2];   // [E]
    const float* basis1 = (const float*)d_in[3];
    const float* comp1  = (const float*)d_in[4];
    const float* root1  = (const float*)d_in[5];
    const float* bias1  = (const float*)d_in[6];
    const float* basis2 = (const float*)d_in[7];
    const float* comp2  = (const float*)d_in[8];
    const float* root2  = (const float*)d_in[9];
    const float* bias2  = (const float*)d_in[10];

    const int N = in_sizes[0] / RGCN_D;            // 50000 (multiple of 16)
    const int E = in_sizes[2];                     // 600000
    const int* src = eidx;
    const int* dst = eidx + E;

    // Workspace layout (floats): W1 | W2 | agg[N,R,128] | cnt[N,R] | h[N,128]
    float* ws  = (float*)d_ws;
    float* W1  = ws;
    float* W2  = W1 + (size_t)RGCN_R * RGCN_DD;
    float* agg = W2 + (size_t)RGCN_R * RGCN_DD;
    float* cnt = agg + (size_t)N * RGCN_R * RGCN_D;
    float* h   = cnt + (size_t)N * RGCN_R;

    const long aggcnt_f = (long)N * RGCN_R * RGCN_D + (long)N * RGCN_R; // contiguous
    const int  wblocks  = (RGCN_R * RGCN_DD + 255) / 256;
    const int  sblocks  = (E * 32 + 255) / 256;     // one wave32 per edge
    const int  tblocks  = (N / 16) * (RGCN_D / 32); // 16-node x 32-col tiles

    // Relation weights for both layers
    rgcn_compute_w<<<wblocks, 256, 0, stream>>>(comp1, basis1, W1);
    rgcn_compute_w<<<wblocks, 256, 0, stream>>>(comp2, basis2, W2);

    // ---- Layer 1 ----
    rgcn_zero_f32<<<2048, 256, 0, stream>>>((float4*)agg, aggcnt_f / 4);
    rgcn_scatter<<<sblocks, 256, 0, stream>>>(x, src, dst, etype, agg, cnt, E);
    rgcn_transform<<<tblocks, 32, 0, stream>>>(x, agg, cnt, W1, root1, bias1, h, 1);

    // ---- Layer 2 ----
    rgcn_zero_f32<<<2048, 256, 0, stream>>>((float4*)agg, aggcnt_f / 4);
    rgcn_scatter<<<sblocks, 256, 0, stream>>>(h, src, dst, etype, agg, cnt, E);
    rgcn_transform<<<tblocks, 32, 0, stream>>>(h, agg, cnt, W2, root2, bias2,
                                               (float*)d_out, 0);
}